// HLLayer_34239479283729
// MI455X (gfx1250) — compile-verified
//
#include <hip/hip_runtime.h>

typedef float v2f __attribute__((ext_vector_type(2)));
typedef float v8f __attribute__((ext_vector_type(8)));

#define NNODES 50000
#define NEDGES 800000
#define IN_DIM 256
#define OUT_DIM 64
#define HEADS 4
#define HD 256          /* HEADS*OUT_DIM */
#define P_L 0.1f

// Guaranteed-native no-return f32 atomic add (GLOBAL_ATOMIC_ADD_F32,
// tracked by STOREcnt; implicit s_wait_idle at s_endpgm covers completion).
__device__ __forceinline__ void atomic_add_f32_noret(float* p, float v) {
  asm volatile("global_atomic_add_f32 %0, %1, off"
               :: "v"(p), "v"(v)
               : "memory");
}

// ---------------------------------------------------------------------------
// K0: fold W_att/W_g into per-input-dim gate vectors.
//   u_d[h*64+k] = sum_e W_att[h][k][e] * W_g[h][e]        (dst half)
//   u_s[h*64+k] = sum_e W_att[h][k][e] * W_g[h][64+e]     (src half)
//   bge[h]      = b_g[h] + sum_e b_att[h][e]*(W_g[h][e]+W_g[h][64+e])
// ---------------------------------------------------------------------------
__global__ __launch_bounds__(256) void k0_fold(
    const float* __restrict__ W_att, const float* __restrict__ b_att,
    const float* __restrict__ W_g,   const float* __restrict__ b_g,
    float* __restrict__ ud, float* __restrict__ us, float* __restrict__ bge) {
  const int j = threadIdx.x;        // 0..255
  const int h = j >> 6;
  const int k = j & 63;
  const float* Wa  = W_att + (h * OUT_DIM + k) * OUT_DIM;
  const float* Wgd = W_g + h * 2 * OUT_DIM;
  const float* Wgs = Wgd + OUT_DIM;
  float sd = 0.f, ss = 0.f;
  for (int e = 0; e < OUT_DIM; ++e) {
    float w = Wa[e];
    sd += w * Wgd[e];
    ss += w * Wgs[e];
  }
  ud[j] = sd;
  us[j] = ss;
  if (j < HEADS) {
    float c = b_g[j];
    const float* ba  = b_att + j * OUT_DIM;
    const float* wgd = W_g + j * 2 * OUT_DIM;
    for (int e = 0; e < OUT_DIM; ++e)
      c += ba[e] * (wgd[e] + wgd[OUT_DIM + e]);
    bge[j] = c;
  }
}

// ---------------------------------------------------------------------------
// K1: hh = relu(h @ W_in + b_in), fp32 WMMA 16x16x4.
// Block = 128 threads = 4 waves. Block owns 16 rows; wave owns 64 cols
// (4 col-tiles). grid = N/16 (50000 % 16 == 0).
// ---------------------------------------------------------------------------
__global__ __launch_bounds__(128) void k1_node_transform(
    const float* __restrict__ h, const float* __restrict__ W_in,
    const float* __restrict__ b_in, float* __restrict__ hh) {
  const int lane = threadIdx.x & 31;
  const int wave = threadIdx.x >> 5;       // 0..3
  const int r    = lane & 15;
  const int kb   = (lane >> 4) << 1;       // 0 or 2
  const int row0 = blockIdx.x * 16;

  v8f c[4] = {};
  const float* arow = h + (size_t)(row0 + r) * IN_DIM + kb;
  const int cb0 = wave * 64 + r;

  for (int kk = 0; kk < IN_DIM / 4; ++kk) {
    v2f a;
    a.x = arow[kk * 4 + 0];
    a.y = arow[kk * 4 + 1];
    const float* bptr = W_in + (size_t)(kk * 4 + kb) * HD + cb0;
#pragma unroll
    for (int ct = 0; ct < 4; ++ct) {
      v2f b;
      b.x = bptr[ct * 16];
      b.y = bptr[ct * 16 + HD];
      c[ct] = __builtin_amdgcn_wmma_f32_16x16x4_f32(
          false, a, false, b, (short)0, c[ct], false, false);
    }
  }

  const int mhi = (lane >> 4) * 8;         // 0 or 8
#pragma unroll
  for (int ct = 0; ct < 4; ++ct) {
    const int col  = wave * 64 + ct * 16 + r;
    const float bs = b_in[col];
    float* op = hh + (size_t)(row0 + mhi) * HD + col;
#pragma unroll
    for (int v = 0; v < 8; ++v) {
      float val = c[ct][v] + bs;
      op[(size_t)v * HD] = val > 0.f ? val : 0.f;
    }
  }
}

// ---------------------------------------------------------------------------
// K2: per-node gate scalars gd/gs via folded vectors; also zero z.
// One wave per node; lane covers 8 consecutive dims (one head per 8 lanes).
// ---------------------------------------------------------------------------
__global__ __launch_bounds__(256) void k2_gates(
    const float* __restrict__ hh, const float* __restrict__ ud,
    const float* __restrict__ us, float* __restrict__ gd,
    float* __restrict__ gs, float* __restrict__ z) {
  const int n = (int)((blockIdx.x * (unsigned)blockDim.x + threadIdx.x) >> 5);
  const int lane = threadIdx.x & 31;
  if (n >= NNODES) return;
  const int j = lane * 8;

  const float4* hp  = (const float4*)(hh + (size_t)n * HD + j);
  const float4* udp = (const float4*)(ud + j);
  const float4* usp = (const float4*)(us + j);
  float4 h0 = hp[0], h1 = hp[1];
  float4 u0 = udp[0], u1 = udp[1];
  float4 v0 = usp[0], v1 = usp[1];

  float pd = h0.x * u0.x + h0.y * u0.y + h0.z * u0.z + h0.w * u0.w +
             h1.x * u1.x + h1.y * u1.y + h1.z * u1.z + h1.w * u1.w;
  float ps = h0.x * v0.x + h0.y * v0.y + h0.z * v0.z + h0.w * v0.w +
             h1.x * v1.x + h1.y * v1.y + h1.z * v1.z + h1.w * v1.w;

#pragma unroll
  for (int m = 1; m < 8; m <<= 1) {
    pd += __shfl_xor(pd, m, 32);
    ps += __shfl_xor(ps, m, 32);
  }
  if ((lane & 7) == 0) {
    const int head = lane >> 3;
    gd[n * HEADS + head] = pd;
    gs[n * HEADS + head] = ps;
  }
  // zero the accumulation buffer for K3
  float4* zp = (float4*)(z + (size_t)n * HD + j);
  zp[0] = make_float4(0.f, 0.f, 0.f, 0.f);
  zp[1] = make_float4(0.f, 0.f, 0.f, 0.f);
}

// ---------------------------------------------------------------------------
// K3: edge scatter. One wave per edge. lane covers 8 dims; head = lane>>3.
// alpha = leaky(gd[dst]+gs[src]+bge) * d[dst]*d[src]; z[dst] += hh[src]*alpha
// via no-return GLOBAL_ATOMIC_ADD_F32 (L2-resident accumulation buffer).
// ---------------------------------------------------------------------------
__global__ __launch_bounds__(256) void k3_edges(
    const int* __restrict__ src, const int* __restrict__ dst,
    const float* __restrict__ d, const float* __restrict__ gd,
    const float* __restrict__ gs, const float* __restrict__ bge,
    const float* __restrict__ hh, float* __restrict__ z) {
  const int e = (int)((blockIdx.x * (unsigned)blockDim.x + threadIdx.x) >> 5);
  if (e >= NEDGES) return;
  const int lane = threadIdx.x & 31;
  const int s = src[e];
  const int t = dst[e];
  const int head = lane >> 3;

  float _h = gd[t * HEADS + head] + gs[s * HEADS + head] + bge[head];
  float g  = _h > 0.f ? _h : -P_L * _h;
  float alpha = g * d[t] * d[s];

  const int j = lane * 8;
  const float4* hp = (const float4*)(hh + (size_t)s * HD + j);
  float4 m0 = hp[0], m1 = hp[1];
  float* zp = z + (size_t)t * HD + j;

  atomic_add_f32_noret(zp + 0, m0.x * alpha);
  atomic_add_f32_noret(zp + 1, m0.y * alpha);
  atomic_add_f32_noret(zp + 2, m0.z * alpha);
  atomic_add_f32_noret(zp + 3, m0.w * alpha);
  atomic_add_f32_noret(zp + 4, m1.x * alpha);
  atomic_add_f32_noret(zp + 5, m1.y * alpha);
  atomic_add_f32_noret(zp + 6, m1.z * alpha);
  atomic_add_f32_noret(zp + 7, m1.w * alpha);
}

// ---------------------------------------------------------------------------
// K4: out = z @ W_out + b_out, fp32 WMMA. Block = 128 threads = 4 waves;
// each wave owns one 16-col tile (64 cols total), block owns 16 rows.
// ---------------------------------------------------------------------------
__global__ __launch_bounds__(128) void k4_out(
    const float* __restrict__ z, const float* __restrict__ W_out,
    const float* __restrict__ b_out, float* __restrict__ out) {
  const int lane = threadIdx.x & 31;
  const int wave = threadIdx.x >> 5;
  const int r    = lane & 15;
  const int kb   = (lane >> 4) << 1;
  const int row0 = blockIdx.x * 16;

  v8f c = {};
  const float* arow = z + (size_t)(row0 + r) * HD + kb;
  const int col = wave * 16 + r;

  for (int kk = 0; kk < HD / 4; ++kk) {
    v2f a;
    a.x = arow[kk * 4 + 0];
    a.y = arow[kk * 4 + 1];
    v2f b;
    b.x = W_out[(size_t)(kk * 4 + kb) * OUT_DIM + col];
    b.y = W_out[(size_t)(kk * 4 + kb + 1) * OUT_DIM + col];
    c = __builtin_amdgcn_wmma_f32_16x16x4_f32(
        false, a, false, b, (short)0, c, false, false);
  }

  const int mhi = (lane >> 4) * 8;
  const float bs = b_out[col];
  float* op = out + (size_t)(row0 + mhi) * OUT_DIM + col;
#pragma unroll
  for (int v = 0; v < 8; ++v)
    op[(size_t)v * OUT_DIM] = c[v] + bs;
}

// ---------------------------------------------------------------------------
extern "C" void kernel_launch(void* const* d_in, const int* in_sizes, int n_in,
                              void* d_out, int out_size, void* d_ws, size_t ws_size,
                              hipStream_t stream) {
  const float* h     = (const float*)d_in[0];
  const int*   src   = (const int*)d_in[1];
  const int*   dst   = (const int*)d_in[2];
  const float* d     = (const float*)d_in[3];
  const float* W_in  = (const float*)d_in[4];
  const float* b_in  = (const float*)d_in[5];
  const float* W_att = (const float*)d_in[6];
  const float* b_att = (const float*)d_in[7];
  const float* W_g   = (const float*)d_in[8];
  const float* b_g   = (const float*)d_in[9];
  const float* W_out = (const float*)d_in[10];
  const float* b_out = (const float*)d_in[11];
  float* out = (float*)d_out;

  float* hh  = (float*)d_ws;                         // N*256 floats (51.2 MB)
  float* z   = hh + (size_t)NNODES * HD;             // N*256 floats (51.2 MB)
  float* gd  = z + (size_t)NNODES * HD;              // N*4
  float* gs  = gd + (size_t)NNODES * HEADS;          // N*4
  float* ud  = gs + (size_t)NNODES * HEADS;          // 256
  float* us  = ud + HD;                              // 256
  float* bge = us + HD;                              // 4

  k0_fold<<<1, 256, 0, stream>>>(W_att, b_att, W_g, b_g, ud, us, bge);
  k1_node_transform<<<NNODES / 16, 128, 0, stream>>>(h, W_in, b_in, hh);
  k2_gates<<<(NNODES * 32) / 256, 256, 0, stream>>>(hh, ud, us, gd, gs, z);
  k3_edges<<<(NEDGES * 32) / 256, 256, 0, stream>>>(src, dst, d, gd, gs, bge, hh, z);
  k4_out<<<NNODES / 16, 128, 0, stream>>>(z, W_out, b_out, out);
}